// InsuranceCycleDetector_34522947125966
// MI455X (gfx1250) — compile-verified
//
#include <hip/hip_runtime.h>
#include <math.h>

// ---------------------------------------------------------------------------
// InsuranceCycleDetector on MI455X (gfx1250)
//
// reference = 4-bin DFT of input_sequence[:,:,0] -> mag/phase -> (128x8)@(8x256)
// The heavy part (claims(128x8192) @ dft_basis(8192x8)) runs on the matrix
// pipe as V_WMMA_F32_16X16X4_F32 with f32 accumulation. HBM-bound: 64 MB of
// strided input (one useful float per 64B line) ~2.7us at 23.3 TB/s; the 1 GB
// hidden_states input is unused by the reference and never touched.
// Twiddles come from the native v_cos_f32 (exact 2*pi*m/8192 arguments,
// -sin(t) = cos(t + pi/2)) -> no libm Payne-Hanek, no EXEC divergence.
// ---------------------------------------------------------------------------

typedef __attribute__((ext_vector_type(2))) float v2f;
typedef __attribute__((ext_vector_type(8))) float v8f;

constexpr int BATCH = 128;
constexpr int SEQ   = 8192;
constexpr int NFEAT = 16;
constexpr int DOUT  = 256;

// np.fft.fftfreq(8192, 1/30); argmin |freq - c| for c in [1/12, 1/4, 1/2, 1]
// (step = 30/8192) -> bins 23, 68, 137, 273
__device__ __forceinline__ int bin_of_pair(int p) {
    return (p == 0) ? 23 : (p == 1) ? 68 : (p == 2) ? 137 : 273;
}

// ---------------------------------------------------------------------------
// Kernel 0: zero the 128x16 f32 bin accumulator in workspace
// ---------------------------------------------------------------------------
__global__ void zero_ws_kernel(float* __restrict__ p, int n) {
    int i = blockIdx.x * blockDim.x + threadIdx.x;
    if (i < n) p[i] = 0.0f;
}

// ---------------------------------------------------------------------------
// Kernel 1: DFT-as-matmul via V_WMMA_F32_16X16X4_F32.
//
// Wave tile: C(16 batches x 16 cols); col j=2k -> Re(bin k) (cos),
// j=2k+1 -> Im(bin k) (-sin), j>=8 -> zero padding.
// Grid: 256 blocks x 256 threads (8 waves) = 2048 waves. Wave w of block
// covers batch_tile = blockIdx.x/32, K-chunk = (blockIdx.x%32)*8 + w; each
// chunk = 32 n-values = 8 WMMA steps, 2 independent cacheline loads per step.
//
// Register layouts (CDNA5 ISA 7.12.2):
//   A 16x4 f32 : vgpr0 = K 2*half, vgpr1 = K 2*half+1; lane%16 = M
//   C/D 16x16  : vgpr v = row (v + 8*half), lane%16 = N
//   B 4x16 f32 : mirrored from A (vgpr = K&1, half = K>>1), lane%16 = N
// ---------------------------------------------------------------------------
__global__ void dft_wmma_kernel(const float* __restrict__ in,   // (128, 8192, 16)
                                float* __restrict__ bins_acc) { // (128, 16) f32
    const int lane  = threadIdx.x & 31;
    const int wave  = threadIdx.x >> 5;
    const int half  = lane >> 4;   // 0/1
    const int lrow  = lane & 15;   // M for A rows, N for B/C cols

    const int batch_tile = blockIdx.x >> 5;                  // 0..7
    const int chunk      = ((blockIdx.x & 31) << 3) | wave;  // 0..255
    const int n0         = chunk * 32;                       // 32 n per chunk

    const long a_row = (long)(batch_tile * 16 + lrow) * SEQ; // claims row base

    // This lane's B column: exact twiddle phase index m = (bin*k) mod 8192,
    // advanced incrementally; angle = 2*pi*m/8192 (+ pi/2 for -sin columns).
    const int   bin        = bin_of_pair(lrow >> 1);
    const bool  is_im      = (lrow & 1) != 0;
    const bool  active_col = (lrow < 8);
    const float PHASE_SCL  = 6.283185307179586f / (float)SEQ;
    const float phase_off  = is_im ? 1.5707963267948966f : 0.0f; // +pi/2 => -sin
    const int   mstep      = (4 * bin) & (SEQ - 1);

    const int k0_first = n0 + 2 * half;                  // K held in vgpr0
    int m0 = (bin * k0_first) & (SEQ - 1);               // bin*k0 mod 8192
    int m1 = (m0 + bin) & (SEQ - 1);                     // bin*(k0+1) mod 8192

    v8f c = {};
    #pragma unroll 4
    for (int s = 0; s < 8; ++s) {
        const int k0 = n0 + (s << 2) + 2 * half;

        // A: single-pass strided gather of claims -> non-temporal
        v2f a;
        a.x = __builtin_nontemporal_load(&in[(a_row + k0) * NFEAT]);
        a.y = __builtin_nontemporal_load(&in[(a_row + k0 + 1) * NFEAT]);

        // B: one native v_cos per element, branchless zeroing of pad columns
        float b0 = __cosf((float)m0 * PHASE_SCL + phase_off);
        float b1 = __cosf((float)m1 * PHASE_SCL + phase_off);
        b0 = active_col ? b0 : 0.0f;
        b1 = active_col ? b1 : 0.0f;
        v2f b; b.x = b0; b.y = b1;

        m0 = (m0 + mstep) & (SEQ - 1);
        m1 = (m1 + mstep) & (SEQ - 1);

        // D = A x B + C  (f32 x f32 -> f32 accumulate on the matrix pipe)
        c = __builtin_amdgcn_wmma_f32_16x16x4_f32(
                /*neg_a=*/false, a, /*neg_b=*/false, b,
                /*c_mod=*/(short)0, c, /*reuse_a=*/false, /*reuse_b=*/false);
    }

    // Reduce partial tiles across K-chunks: C vgpr v = batch row v + 8*half
    #pragma unroll
    for (int v = 0; v < 8; ++v) {
        const int bm = batch_tile * 16 + v + 8 * half;
        atomicAdd(&bins_acc[bm * 16 + lrow], c[v]);
    }
}

// ---------------------------------------------------------------------------
// Kernel 2: mag/phase, cycle weighting, (128x8)@(8x256) projection + bias
// ---------------------------------------------------------------------------
__global__ void project_kernel(const float* __restrict__ bins_acc, // (128,16)
                               const float* __restrict__ cw,       // (4,)
                               const float* __restrict__ pw,       // (256,8)
                               const float* __restrict__ pb,       // (256,)
                               float* __restrict__ out) {          // (128,256)
    __shared__ float sf[8];
    const int b = blockIdx.x;
    const int d = threadIdx.x;

    if (d < 4) {
        const float re  = bins_acc[b * 16 + 2 * d];
        const float im  = bins_acc[b * 16 + 2 * d + 1];
        const float mag = sqrtf(re * re + im * im);
        const float ph  = atan2f(im, re);   // accurate: phase feeds the output
        const float w   = cw[d];
        sf[2 * d]     = mag * w;   // feats interleaved [m0,p0,m1,p1,...]
        sf[2 * d + 1] = ph * w;
    }
    __syncthreads();

    float acc = pb[d];
    #pragma unroll
    for (int j = 0; j < 8; ++j) acc += sf[j] * pw[d * 8 + j];
    out[b * DOUT + d] = acc;
}

// ---------------------------------------------------------------------------
extern "C" void kernel_launch(void* const* d_in, const int* in_sizes, int n_in,
                              void* d_out, int out_size, void* d_ws, size_t ws_size,
                              hipStream_t stream) {
    (void)in_sizes; (void)n_in; (void)out_size; (void)ws_size;

    const float* in_seq = (const float*)d_in[0];   // (128, 8192, 16)
    // d_in[1] = hidden_states: unused by the reference -> never read (1 GB saved)
    const float* cw     = (const float*)d_in[2];   // (4,)
    const float* pw     = (const float*)d_in[3];   // (256, 8)
    const float* pb     = (const float*)d_in[4];   // (256,)
    float*       out    = (float*)d_out;           // (128, 256)
    float*       acc    = (float*)d_ws;            // 128*16 f32 accumulator

    const int acc_n = BATCH * 16;
    zero_ws_kernel<<<(acc_n + 255) / 256, 256, 0, stream>>>(acc, acc_n);
    dft_wmma_kernel<<<256, 256, 0, stream>>>(in_seq, acc);
    project_kernel<<<BATCH, DOUT, 0, stream>>>(acc, cw, pw, pb, out);
}